// _PCEN_75033078661345
// MI455X (gfx1250) — compile-verified
//
#include <hip/hip_runtime.h>

#define AS1 __attribute__((address_space(1)))
#define AS3 __attribute__((address_space(3)))

typedef int v4i __attribute__((vector_size(16)));

#define T_LEN      8000
#define CHUNK      32
#define NCHUNK     250          // 250 * 32 = 8000
#define B128_UNITS 2000         // 8000 floats / 4 per 16B packet
#define NROWS      4096         // B*C*F = 32*1*128
#define FLOOR_V    1e-6f

// ---- CDNA5 async global<->LDS movers (ASYNCcnt-tracked) ------------------

__device__ __forceinline__ void async_load_b128(const float* g, float* l) {
#if __has_builtin(__builtin_amdgcn_global_load_async_to_lds_b128)
  __builtin_amdgcn_global_load_async_to_lds_b128(
      (AS1 v4i*)(v4i*)(void*)g, (AS3 v4i*)(v4i*)(void*)l, 0, 0);
#else
  AS3 float* l3 = (AS3 float*)(void*)l;
  unsigned lds_off = (unsigned)(unsigned long)l3;
  asm volatile("global_load_async_to_lds_b128 %0, %1, off"
               :: "v"(lds_off), "v"((const void*)g) : "memory");
#endif
}

__device__ __forceinline__ void async_store_b128(float* g, const float* l) {
#if __has_builtin(__builtin_amdgcn_global_store_async_from_lds_b128)
  __builtin_amdgcn_global_store_async_from_lds_b128(
      (AS1 v4i*)(v4i*)(void*)g, (AS3 v4i*)(v4i*)(void*)l, 0, 0);
#else
  AS3 const float* l3 = (AS3 const float*)(const void*)l;
  unsigned lds_off = (unsigned)(unsigned long)l3;
  asm volatile("global_store_async_from_lds_b128 %0, %1, off"
               :: "v"(lds_off), "v"((void*)g) : "memory");
#endif
}

__device__ __forceinline__ void wait_async0() {
#if __has_builtin(__builtin_amdgcn_s_wait_asynccnt)
  __builtin_amdgcn_s_wait_asynccnt(0);
#else
  asm volatile("s_wait_asynccnt 0" ::: "memory");
#endif
}

// ---- PCEN: one row (length 8000) per 256-thread workgroup ----------------

__global__ __launch_bounds__(256) void _PCEN_75033078661345_kernel(
    const float* __restrict__ mag,   const float* __restrict__ alpha,
    const float* __restrict__ delta, const float* __restrict__ root,
    const float* __restrict__ smooth,
    float* __restrict__ out, float* __restrict__ hidden)
{
  __shared__ __align__(16) float row[T_LEN];
  __shared__ float sP[8], sQ[8];

  const int r    = blockIdx.x;
  const int tid  = threadIdx.x;
  const int lane = tid & 31;
  const int wv   = tid >> 5;

  const float* gin  = mag + (size_t)r * T_LEN;
  float*       gout = out + (size_t)r * T_LEN;

  // ---- Phase A: async copy row -> LDS (coalesced B128) -------------------
  for (int i = tid; i < B128_UNITS; i += 256)
    async_load_b128(gin + i * 4, row + i * 4);
  wait_async0();          // this wave's async writes to LDS are done
  __syncthreads();        // all waves' writes visible

  // ---- Parameters (uniform; C == 1) --------------------------------------
  float w = smooth[0]; w = w < 0.f ? 0.f : (w > 1.f ? 1.f : w);
  const float omw = 1.0f - w;
  float a = alpha[0];  a = a < 1.f ? a : 1.f;
  const float d = delta[0];
  float rt = root[0];  rt = rt > 1.f ? rt : 1.f;
  const float inv_r = 1.0f / rt;
  const float drt   = powf(d, inv_r);
  const float x0    = row[0];   // scan init == x[0]; ema[0] == x[0]

  // ---- Phase B: per-chunk affine reduction acc_out = P*acc_in + Q --------
  float P = 1.0f, Q = 0.0f;
  if (tid < NCHUNK) {
    const float* xp = row + tid * CHUNK;
#pragma unroll 8
    for (int i = 0; i < CHUNK; ++i) {
      Q = w * xp[i] + omw * Q;
      P *= omw;
    }
  }

  // ---- Phase C1: wave32 inclusive scan of affine maps (shfl_up) ----------
#pragma unroll
  for (int dlt = 1; dlt < 32; dlt <<= 1) {
    float Pp = __shfl_up(P, dlt, 32);
    float Qp = __shfl_up(Q, dlt, 32);
    if (lane >= dlt) { Q = P * Qp + Q; P = P * Pp; }
  }
  if (lane == 31) { sP[wv] = P; sQ[wv] = Q; }
  __syncthreads();

  // ---- Phase C2: exclusive scan of the 8 wave aggregates -----------------
  if (tid == 0) {
    float p = 1.0f, q = 0.0f;
#pragma unroll
    for (int i = 0; i < 8; ++i) {
      float tp = sP[i], tq = sQ[i];
      sP[i] = p; sQ[i] = q;
      q = tp * q + tq;
      p = tp * p;
    }
  }
  __syncthreads();

  // ---- Phase C3: per-thread exclusive prefix -> incoming EMA state -------
  float Pl = __shfl_up(P, 1, 32);
  float Ql = __shfl_up(Q, 1, 32);
  if (lane == 0) { Pl = 1.0f; Ql = 0.0f; }
  const float Pw = sP[wv], Qw = sQ[wv];
  const float Pex = Pl * Pw;
  const float Qex = Pl * Qw + Ql;
  float acc = Pex * x0 + Qex;

  // ---- Phase D: replay recurrence in-chunk + pointwise PCEN, in place ----
  if (tid < NCHUNK) {
    float* xp = row + tid * CHUNK;
#pragma unroll 4
    for (int i = 0; i < CHUNK; ++i) {
      float x = xp[i];
      acc = w * x + omw * acc;                  // ema
      float den = powf(FLOOR_V + acc, a);
      float v   = x / den + d;
      xp[i] = powf(v, inv_r) - drt;
    }
    if (tid == NCHUNK - 1) hidden[r] = acc;     // final state -> hidden_state
  }
  __syncthreads();

  // ---- Phase E: async store LDS -> out (coalesced B128) ------------------
  for (int i = tid; i < B128_UNITS; i += 256)
    async_store_b128(gout + i * 4, row + i * 4);
  wait_async0();
}

extern "C" void kernel_launch(void* const* d_in, const int* in_sizes, int n_in,
                              void* d_out, int out_size, void* d_ws, size_t ws_size,
                              hipStream_t stream) {
  const float* mag    = (const float*)d_in[0];
  const float* alpha  = (const float*)d_in[1];
  const float* delta  = (const float*)d_in[2];
  const float* root   = (const float*)d_in[3];
  const float* smooth = (const float*)d_in[4];
  float* out    = (float*)d_out;
  float* hidden = out + (size_t)NROWS * T_LEN;   // outputs concatenated flat

  _PCEN_75033078661345_kernel<<<NROWS, 256, 0, stream>>>(
      mag, alpha, delta, root, smooth, out, hidden);
}